// NeuralGeodesicFlows_45784351375900
// MI455X (gfx1250) — compile-verified
//
#include <hip/hip_runtime.h>

// CDNA5 (gfx1250) wave32 WMMA f32 16x16x4 implementation of neural geodesic flow RK4.

typedef __attribute__((ext_vector_type(2))) float v2f;
typedef __attribute__((ext_vector_type(4))) float v4f;
typedef __attribute__((ext_vector_type(8))) float v8f;

#define WMMA_F32(a, b, c) \
  __builtin_amdgcn_wmma_f32_16x16x4_f32(false, (a), false, (b), (short)0, (c), false, false)

// Branch-free tanh: tanh(x) = (e-1)/(e+1), e = exp2(2*log2(e)*clamp(x)).
// Keeps EXEC all-ones (no divergent slow-path like libm tanhf).
__device__ __forceinline__ float tanh_fast(float x) {
  float xc = fminf(fmaxf(x, -9.0f), 9.0f);
  float e  = __builtin_amdgcn_exp2f(xc * 2.8853900817779268f);  // 2/ln(2)
  return (e - 1.0f) * __builtin_amdgcn_rcpf(e + 1.0f);
}

// One RHS evaluation (geodesic ODE f) for the wave's 16 points.
// zin/kout: LDS, 16 points x 32 floats ([x(16) | v(16)]).
// W1T_l: LDS 64x16 (W1 transposed), W1rm_l: LDS 16x64 (row-major).
template <bool USE_W2T>
__device__ __forceinline__ void geo_feval(
    const float* __restrict__ W2,    // global, 64x256 row-major
    const float* __restrict__ W2T,   // global scratch, 256x64 (valid if USE_W2T)
    const float* W1T_l, const float* W1rm_l,
    const float* b1_l, const float* b2_l,
    const float* zin, float* kout,
    float* Araw, float* Hbuf, float* wbuf, float* sbuf,
    int h, int n)
{
  // ---------------- GEMM1: U = X @ W1 + b1 ; H = tanh(U) -> Hbuf(16x64) ----
  // 4 independent accumulator chains (one per 16-col tile), K round-robin.
  {
    v2f xfrag[4];
#pragma unroll
    for (int kc = 0; kc < 4; ++kc)
      xfrag[kc] = *(const v2f*)&zin[n * 32 + 4 * kc + 2 * h];   // x part of z
    v8f c[4];
#pragma unroll
    for (int nt = 0; nt < 4; ++nt) {
      float bias = b1_l[16 * nt + n];
#pragma unroll
      for (int r = 0; r < 8; ++r) c[nt][r] = bias;
    }
#pragma unroll
    for (int kc = 0; kc < 4; ++kc) {
#pragma unroll
      for (int nt = 0; nt < 4; ++nt) {
        v2f bf = *(const v2f*)&W1T_l[(16 * nt + n) * 16 + 4 * kc + 2 * h];
        c[nt] = WMMA_F32(xfrag[kc], bf, c[nt]);
      }
    }
#pragma unroll
    for (int nt = 0; nt < 4; ++nt)
#pragma unroll
      for (int r = 0; r < 8; ++r)
        Hbuf[(r + 8 * h) * 64 + 16 * nt + n] = tanh_fast(c[nt][r]);
  }
  __syncthreads();

  // ---------------- GEMM2: Araw = H @ W2 + b2 (16x256) --------------------
  // Tiles processed in groups of 4: 4 independent chains, A-frag reused 4x.
  v2f hfrag[16];
#pragma unroll
  for (int kc = 0; kc < 16; ++kc)
    hfrag[kc] = *(const v2f*)&Hbuf[n * 64 + 4 * kc + 2 * h];
#pragma unroll
  for (int ntg = 0; ntg < 4; ++ntg) {
    v8f c[4];
#pragma unroll
    for (int nt4 = 0; nt4 < 4; ++nt4) {
      float bias = b2_l[16 * (4 * ntg + nt4) + n];
#pragma unroll
      for (int r = 0; r < 8; ++r) c[nt4][r] = bias;
    }
#pragma unroll
    for (int kc = 0; kc < 16; ++kc) {
      int krow = 4 * kc + 2 * h;
#pragma unroll
      for (int nt4 = 0; nt4 < 4; ++nt4) {
        int nt = 4 * ntg + nt4;
        v2f bf;
        if (USE_W2T) {
          bf = *(const v2f*)(W2T + (16 * nt + n) * 64 + krow);
        } else {
          bf.x = W2[krow * 256 + 16 * nt + n];
          bf.y = W2[(krow + 1) * 256 + 16 * nt + n];
        }
        c[nt4] = WMMA_F32(hfrag[kc], bf, c[nt4]);
      }
    }
#pragma unroll
    for (int nt4 = 0; nt4 < 4; ++nt4)
#pragma unroll
      for (int r = 0; r < 8; ++r)
        Araw[(r + 8 * h) * 256 + 16 * (4 * ntg + nt4) + n] = c[nt4][r];
  }
  __syncthreads();

  // ---------------- per point: w = A^T v  (16 lanes/point, 2 points/pass) --
#pragma unroll 1
  for (int pp = 0; pp < 8; ++pp) {
    int p = 2 * pp + h;
    float wv = 0.0f;
#pragma unroll
    for (int a = 0; a < 16; ++a)
      wv += Araw[p * 256 + a * 16 + n] * zin[p * 32 + 16 + a];
    wbuf[p * 16 + n] = wv;
  }
  __syncthreads();

  // ---------------- GEMM3: Q = P @ W2^T, P[p][16a+j] = v_a * w_j ----------
  // A-frag built on the fly (row of P = point index = n); 4 independent
  // accumulator chains over the 4 output tiles, A-frag reused 4x per K step.
  {
    int p = n;
    float vreg[16];
#pragma unroll
    for (int d0 = 0; d0 < 16; ++d0) vreg[d0] = zin[p * 32 + 16 + d0];
    v8f c[4];
#pragma unroll
    for (int nt = 0; nt < 4; ++nt)
#pragma unroll
      for (int r = 0; r < 8; ++r) c[nt][r] = 0.0f;
#pragma unroll
    for (int kc = 0; kc < 64; ++kc) {
      int a = kc >> 2;                       // compile-time per unrolled kc
      v2f wv2 = *(const v2f*)&wbuf[p * 16 + (kc & 3) * 4 + 2 * h];
      v2f af;
      af.x = vreg[a] * wv2.x;
      af.y = vreg[a] * wv2.y;
#pragma unroll
      for (int nt = 0; nt < 4; ++nt) {
        v2f bf = *(const v2f*)(W2 + (16 * nt + n) * 256 + 4 * kc + 2 * h);
        c[nt] = WMMA_F32(af, bf, c[nt]);
      }
    }
    // fused: R = (1 - H^2) .* Q, written back into Hbuf (same-lane elements)
#pragma unroll
    for (int nt = 0; nt < 4; ++nt)
#pragma unroll
      for (int r = 0; r < 8; ++r) {
        int row = r + 8 * h, col = 16 * nt + n;
        float hv = Hbuf[row * 64 + col];
        Hbuf[row * 64 + col] = c[nt][r] * (1.0f - hv * hv);
      }
  }
  __syncthreads();

  // ---------------- GEMM4: S = R @ W1^T (16x16) -> sbuf -------------------
  {
    v8f c;
#pragma unroll
    for (int r = 0; r < 8; ++r) c[r] = 0.0f;
#pragma unroll
    for (int kc = 0; kc < 16; ++kc) {
      v2f af = *(const v2f*)&Hbuf[n * 64 + 4 * kc + 2 * h];
      v2f bf = *(const v2f*)&W1rm_l[n * 64 + 4 * kc + 2 * h];
      c = WMMA_F32(af, bf, c);
    }
#pragma unroll
    for (int r = 0; r < 8; ++r)
      sbuf[(r + 8 * h) * 16 + n] = c[r];
  }
  __syncthreads();

  // ---------------- per point: g = A A^T + I ; solve g y = s ; dv = -y ----
#pragma unroll 1
  for (int pp = 0; pp < 8; ++pp) {
    int p = 2 * pp + h;
    int i = n;                                  // this lane owns row i
    float ar[16];
#pragma unroll
    for (int m = 0; m < 16; ++m) ar[m] = Araw[p * 256 + i * 16 + m];
    float g[16];
#pragma unroll
    for (int j = 0; j < 16; ++j) {
      float acc = (i == j) ? 1.0f : 0.0f;
#pragma unroll
      for (int m = 0; m < 16; ++m)
        acc += ar[m] * Araw[p * 256 + j * 16 + m];
      g[j] = acc;
    }
    float s = sbuf[p * 16 + i];
    float diag = 1.0f;
    // Gauss-Jordan elimination, no pivoting (g is SPD, diag >= 1).
    // Width-16 shuffles keep the two points in the wave independent.
#pragma unroll
    for (int k = 0; k < 16; ++k) {
      float piv = __shfl(g[k], k, 16);
      diag = (i == k) ? g[k] : diag;            // g[i][i] is final at step k==i
      float f = (i == k) ? 0.0f : (g[k] / piv); // masked update factor
#pragma unroll
      for (int j = k + 1; j < 16; ++j) {
        float gkj = __shfl(g[j], k, 16);
        g[j] -= f * gkj;
      }
      float sk = __shfl(s, k, 16);
      s -= f * sk;
    }
    float dv = -(s / diag);
    kout[p * 32 + i] = zin[p * 32 + 16 + i];    // dx = v
    kout[p * 32 + 16 + i] = dv;
  }
  __syncthreads();
}

template <bool USE_W2T>
__global__ void __launch_bounds__(32, 1)
geoflow_kernel(const float* __restrict__ z,
               const float* __restrict__ tptr,
               const float* __restrict__ W1,   // 16x64 row-major
               const float* __restrict__ b1,   // 64
               const float* __restrict__ W2,   // 64x256 row-major
               const float* __restrict__ b2,   // 256
               const int* __restrict__ nsptr,
               const float* __restrict__ W2T,  // 256x64 (d_ws), valid if USE_W2T
               float* __restrict__ out)
{
  __shared__ __align__(16) float Araw[16 * 256];
  __shared__ __align__(16) float Hbuf[16 * 64];
  __shared__ __align__(16) float wbuf[16 * 16];
  __shared__ __align__(16) float sbuf[16 * 16];
  __shared__ __align__(16) float zcur[512];
  __shared__ __align__(16) float ztmp[512];
  __shared__ __align__(16) float kbuf[512];
  __shared__ __align__(16) float kacc[512];
  __shared__ __align__(16) float W1T_l[64 * 16];   // W1T[m][d] = W1[d][m]
  __shared__ __align__(16) float W1rm_l[16 * 64];  // row-major copy
  __shared__ __align__(16) float b1_l[64];
  __shared__ __align__(16) float b2_l[256];

  int lane = (int)threadIdx.x & 31;
  int h = lane >> 4, n = lane & 15;

  // Stage W1 (both orders), b1, b2 into LDS (keeps VGPR pressure low: all
  // B-fragments are ds_load_b64 at point of use instead of persistent regs).
#pragma unroll
  for (int q = 0; q < 8; ++q) {
    int idx = lane * 8 + q;
#pragma unroll
    for (int sweep = 0; sweep < 4; ++sweep) {
      int e = idx + sweep * 256;
      W1rm_l[e] = W1[e];
      int c = e >> 4, r = e & 15;
      W1T_l[e] = W1[r * 64 + c];
    }
  }
#pragma unroll
  for (int q = 0; q < 2; ++q) b1_l[lane * 2 + q] = b1[lane * 2 + q];
#pragma unroll
  for (int q = 0; q < 8; ++q) b2_l[lane * 8 + q] = b2[lane * 8 + q];

  // Load this wave's 16 points (512 floats) into LDS.
  size_t base = (size_t)blockIdx.x * 512;
#pragma unroll
  for (int q = 0; q < 4; ++q)
    *(v4f*)&zcur[lane * 16 + 4 * q] = *(const v4f*)(z + base + lane * 16 + 4 * q);
  __syncthreads();

  int NS = nsptr[0];
  float dt = tptr[0] / (float)NS;

  for (int step = 0; step < NS; ++step) {
#pragma unroll 1
    for (int st = 0; st < 4; ++st) {
      const float* zin = (st == 0) ? zcur : ztmp;
      geo_feval<USE_W2T>(W2, W2T, W1T_l, W1rm_l, b1_l, b2_l, zin, kbuf,
                         Araw, Hbuf, wbuf, sbuf, h, n);
      float ac = (st == 0 || st == 3) ? 1.0f : 2.0f;
      float bc = (st < 2) ? (0.5f * dt) : dt;
#pragma unroll
      for (int e = 0; e < 16; ++e) {
        int idx = lane * 16 + e;
        float kv = kbuf[idx];
        kacc[idx] = (st == 0) ? kv : (kacc[idx] + ac * kv);
        if (st < 3) ztmp[idx] = zcur[idx] + bc * kv;
      }
      __syncthreads();
    }
#pragma unroll
    for (int e = 0; e < 16; ++e) {
      int idx = lane * 16 + e;
      zcur[idx] += (dt * (1.0f / 6.0f)) * kacc[idx];
    }
    __syncthreads();
  }

#pragma unroll
  for (int q = 0; q < 4; ++q)
    *(v4f*)(out + base + lane * 16 + 4 * q) = *(const v4f*)&zcur[lane * 16 + 4 * q];
}

// Pre-pass: W2T[c][m] = W2[m][c]  (64x256 -> 256x64), written into d_ws.
extern "C" __global__ void __launch_bounds__(256, 1)
w2_transpose_kernel(const float* __restrict__ W2, float* __restrict__ W2T) {
  int idx = (int)blockIdx.x * 256 + (int)threadIdx.x;  // 16384 elements
  int m = idx >> 8, c = idx & 255;
  W2T[c * 64 + m] = W2[idx];
}

extern "C" void kernel_launch(void* const* d_in, const int* in_sizes, int n_in,
                              void* d_out, int out_size, void* d_ws, size_t ws_size,
                              hipStream_t stream) {
  const float* z  = (const float*)d_in[0];
  const float* t  = (const float*)d_in[1];
  const float* W1 = (const float*)d_in[2];
  const float* b1 = (const float*)d_in[3];
  const float* W2 = (const float*)d_in[4];
  const float* b2 = (const float*)d_in[5];
  const int*   ns = (const int*)d_in[6];
  float* out = (float*)d_out;
  int nblocks = in_sizes[0] / 512;   // 16 points x 32 floats per single-wave block

  if (ws_size >= 64 * 256 * sizeof(float)) {
    float* W2T = (float*)d_ws;
    hipLaunchKernelGGL(w2_transpose_kernel, dim3(64), dim3(256), 0, stream, W2, W2T);
    hipLaunchKernelGGL((geoflow_kernel<true>), dim3(nblocks), dim3(32), 0, stream,
                       z, t, W1, b1, W2, b2, ns, W2T, out);
  } else {
    hipLaunchKernelGGL((geoflow_kernel<false>), dim3(nblocks), dim3(32), 0, stream,
                       z, t, W1, b1, W2, b2, ns, (const float*)nullptr, out);
  }
}